// FasterRCNN_66417374265648
// MI455X (gfx1250) — compile-verified
//
#include <hip/hip_runtime.h>
#include <hip/hip_bf16.h>

// Faster R-CNN bbox decode + 21-way softmax for MI455X (gfx1250).
// Memory-bound: ~428 MB of traffic -> ~18us floor at 23.3 TB/s HBM.
// Strategy: every byte moves via aligned b128 vector memory ops with NT
// temporal hints (touch-once streams, footprint 2.2x the 192MB L2 -> don't
// pollute it); rois (8MB, 21-lane broadcast reuse) stays on the default RT
// path. v_exp_f32 (TRANS) co-executes with packed VALU; global_prefetch_b8
// (speculative, SYS scope) primes the dominant streams. No WMMA: arithmetic
// intensity (~1 flop/byte) is ~2 orders of magnitude below machine balance,
// so the matrix pipes cannot contribute to this workload.

#define NUM_CLASSES 21
#define IMG_H_F 600.0f
#define IMG_W_F 800.0f

typedef float v4f __attribute__((ext_vector_type(4)));  // NT-builtin friendly

// ---------------------------------------------------------------------------
// Kernel 1: per-(row,class) bbox decode.
//   thread t in [0, R*21):  r = t/21, handles 4 floats of loc -> 4 of bbox.
//   loc and bbox are flat v4f[R*21]: aligned, fully coalesced b128, NT.
//   rois[r] is a 16-byte row shared by 21 consecutive lanes (RT: cache hit).
// ---------------------------------------------------------------------------
__global__ __launch_bounds__(256) void frcnn_bbox_decode_kernel(
    const float4* __restrict__ rois,   // R   x (ymin, xmin, ymax, xmax)
    const v4f*    __restrict__ loc,    // R*C x (dy, dx, dh, dw), pre-denorm
    v4f*          __restrict__ bbox,   // R*C x (y0, x0, y1, x1)
    int total)                         // R*C
{
    int t = blockIdx.x * blockDim.x + threadIdx.x;
    if (t >= total) return;

    // Prefetch the dominant stream ~64KB ahead. Lowers to global_prefetch_b8
    // at SYS scope == speculative, so running past the buffer end is safe.
    __builtin_prefetch(&loc[t + 4096], 0, 0);

    int r = t / NUM_CLASSES;           // magic-number multiply, no divide

    v4f    l = __builtin_nontemporal_load(&loc[t]);  // b128 NT, unit-stride
    float4 b = rois[r];                // RT: broadcast within 21-lane run

    // loc2bbox with LOC_STD = (0.1, 0.1, 0.2, 0.2), LOC_MEAN = 0
    float dy = l.x * 0.1f;
    float dx = l.y * 0.1f;
    float dh = l.z * 0.2f;
    float dw = l.w * 0.2f;

    float src_h  = b.z - b.x;
    float src_w  = b.w - b.y;
    float src_cy = fmaf(0.5f, src_h, b.x);
    float src_cx = fmaf(0.5f, src_w, b.y);

    float cy = fmaf(dy, src_h, src_cy);
    float cx = fmaf(dx, src_w, src_cx);
    float h  = __expf(dh) * src_h;     // v_exp_f32 (TRANS, co-executes w/ VALU)
    float w  = __expf(dw) * src_w;

    v4f o;
    o.x = fminf(fmaxf(fmaf(-0.5f, h, cy), 0.0f), IMG_H_F);
    o.y = fminf(fmaxf(fmaf(-0.5f, w, cx), 0.0f), IMG_W_F);
    o.z = fminf(fmaxf(fmaf( 0.5f, h, cy), 0.0f), IMG_H_F);
    o.w = fminf(fmaxf(fmaf( 0.5f, w, cx), 0.0f), IMG_W_F);

    __builtin_nontemporal_store(o, &bbox[t]);        // b128 NT, unit-stride
}

// ---------------------------------------------------------------------------
// Kernel 2: numerically-stable softmax over 21 classes, FOUR rows per thread.
//   4 rows x 21 floats = 84 floats = 336 B = exactly 21 aligned float4s
//   (row base 4k*84B is 16B-aligned), so the whole score/prob stream moves
//   as NT b128 vector ops. All register indexing unrolls to constants, so
//   the 84-float tile lives entirely in VGPRs.
// ---------------------------------------------------------------------------
__global__ __launch_bounds__(256) void frcnn_softmax21x4_kernel(
    const float* __restrict__ scores,  // R x 21
    float* __restrict__ prob,          // R x 21
    int R)
{
    int t  = blockIdx.x * blockDim.x + threadIdx.x;
    int r0 = t * 4;
    if (r0 >= R) return;

    if (r0 + 4 <= R) {
        const v4f* src = (const v4f*)(scores) + (size_t)t * NUM_CLASSES;
        v4f*       dst = (v4f*)(prob)         + (size_t)t * NUM_CLASSES;

        // ~86KB ahead on the score stream (speculative prefetch, safe OOB).
        __builtin_prefetch(src + 256 * NUM_CLASSES, 0, 0);

        v4f f4[NUM_CLASSES];           // 4 rows staged in registers
#pragma unroll
        for (int j = 0; j < NUM_CLASSES; ++j)
            f4[j] = __builtin_nontemporal_load(&src[j]);   // 21x b128 NT

        float* v = (float*)f4;
#pragma unroll
        for (int row = 0; row < 4; ++row) {
            float* x = v + row * NUM_CLASSES;
            float m = x[0];
#pragma unroll
            for (int j = 1; j < NUM_CLASSES; ++j) m = fmaxf(m, x[j]);
            float s = 0.0f;
#pragma unroll
            for (int j = 0; j < NUM_CLASSES; ++j) {
                x[j] = __expf(x[j] - m);   // v_exp_f32 (TRANS)
                s += x[j];
            }
            float inv = 1.0f / s;          // one rcp+refine per row
#pragma unroll
            for (int j = 0; j < NUM_CLASSES; ++j) x[j] *= inv;
        }

#pragma unroll
        for (int j = 0; j < NUM_CLASSES; ++j)
            __builtin_nontemporal_store(f4[j], &dst[j]);   // 21x b128 NT
    } else {
        // Tail (R not divisible by 4): scalar per-row fallback.
        for (int r = r0; r < R; ++r) {
            const float* x = scores + (size_t)r * NUM_CLASSES;
            float*       o = prob   + (size_t)r * NUM_CLASSES;
            float m = x[0];
            for (int j = 1; j < NUM_CLASSES; ++j) m = fmaxf(m, x[j]);
            float e[NUM_CLASSES];
            float s = 0.0f;
            for (int j = 0; j < NUM_CLASSES; ++j) {
                e[j] = __expf(x[j] - m);
                s += e[j];
            }
            float inv = 1.0f / s;
            for (int j = 0; j < NUM_CLASSES; ++j) o[j] = e[j] * inv;
        }
    }
}

// ---------------------------------------------------------------------------
// Launch: inputs are [rois (R*4), roi_cls_loc (R*84), roi_scores (R*21)],
// output is bbox (R*84) followed by prob (R*21), all float32.
// ---------------------------------------------------------------------------
extern "C" void kernel_launch(void* const* d_in, const int* in_sizes, int n_in,
                              void* d_out, int out_size, void* d_ws, size_t ws_size,
                              hipStream_t stream) {
    const float* rois = (const float*)d_in[0];
    const float* loc  = (const float*)d_in[1];
    const float* scr  = (const float*)d_in[2];

    const int R = in_sizes[0] / 4;

    float* bbox = (float*)d_out;                                   // R*84
    float* prob = (float*)d_out + (size_t)R * (NUM_CLASSES * 4);   // R*21

    const int tb = 256;                 // 8 wave32s per block

    const int total = R * NUM_CLASSES;  // 10.5M (r, class) work items
    frcnn_bbox_decode_kernel<<<(total + tb - 1) / tb, tb, 0, stream>>>(
        (const float4*)rois, (const v4f*)loc, (v4f*)bbox, total);

    const int rowThreads = (R + 3) / 4; // 4 rows per thread
    frcnn_softmax21x4_kernel<<<(rowThreads + tb - 1) / tb, tb, 0, stream>>>(
        scr, prob, R);
}